// HeteroDistMultPredictor_45440753992135
// MI455X (gfx1250) — compile-verified
//
#include <hip/hip_runtime.h>

typedef __attribute__((ext_vector_type(2))) float v2f;
typedef __attribute__((ext_vector_type(4))) float v4f;
typedef __attribute__((ext_vector_type(8))) float v8f;

#define DIM 128
#define MAX_REL 8
#define EDGES_PER_WAVE 16
#define WAVES_PER_BLOCK 8
#define EDGES_PER_BLOCK (EDGES_PER_WAVE * WAVES_PER_BLOCK)

// DistMult edge scoring with fp32 WMMA:
//   D(16x16) = A(16x4) x B(4x16) accumulated over K=128 in 32 chunks.
//   A[m,k] = x[src_m][k] * r_emb[rel_m][k], B[k,n] = x[dst_n][k]
//   score of edge m = D[m][m] (diagonal).
__global__ __launch_bounds__(256)
void distmult_wmma_kernel(const float* __restrict__ x,
                          const float* __restrict__ remb,
                          const int* __restrict__ src,
                          const int* __restrict__ dst,
                          const int* __restrict__ rel,
                          float* __restrict__ out,
                          int n_edges, int n_rel)
{
    __shared__ float lds_rel[MAX_REL * DIM];   // 4 KB: all relation embeddings

    // Cooperative stage of the (tiny) relation table into LDS, 16B per thread.
    {
        int t = threadIdx.x;
        int total4 = (n_rel * DIM) >> 2;       // = 256 for N_REL=8
        if (t < total4) {
            ((v4f*)lds_rel)[t] = ((const v4f*)remb)[t];
        }
    }
    __syncthreads();

    const int lane = threadIdx.x & 31;
    const int wave = threadIdx.x >> 5;
    const int l16  = lane & 15;                // edge slot within the group of 16
    const int hi   = lane >> 4;                // 0: K=k0,k0+1   1: K=k0+2,k0+3

    const int eBase = blockIdx.x * EDGES_PER_BLOCK + wave * EDGES_PER_WAVE;
    int e  = eBase + l16;
    int ec = e < n_edges ? e : (n_edges - 1);  // clamp (grid divides exactly anyway)

    const int si = src[ec];
    const int di = dst[ec];
    const int ri = rel[ec];

    // Each lane's 8-byte window into its edge's rows, offset by 2*hi floats.
    const float* __restrict__ srow = x + (size_t)si * DIM + 2 * hi;
    const float* __restrict__ drow = x + (size_t)di * DIM + 2 * hi;
    const float* __restrict__ rrow = lds_rel + ri * DIM + 2 * hi;

    v8f acc = {};

    // 32 x V_WMMA_F32_16X16X4_F32, K advances by 4 each step.
    // EXEC is all-ones here (no divergence) as required by the WMMA spec.
#pragma unroll
    for (int k0 = 0; k0 < DIM; k0 += 4) {
        v2f a = *(const v2f*)(srow + k0);      // global_load_b64 (L2-resident gather)
        v2f b = *(const v2f*)(drow + k0);      // global_load_b64
        v2f r = *(const v2f*)(rrow + k0);      // ds_load_b64
        a.x *= r.x;
        a.y *= r.y;
        // 8 args: (neg_a, A, neg_b, B, c_mod, C, reuse_a, reuse_b)
        acc = __builtin_amdgcn_wmma_f32_16x16x4_f32(
            /*neg_a=*/false, a, /*neg_b=*/false, b,
            /*c_mod=*/(short)0, acc, /*reuse_a=*/false, /*reuse_b=*/false);
    }

    // Diagonal extraction from the 16x16 f32 C/D layout:
    //   lanes 0-15:  N=lane,    vgpr v holds M=v      -> diag m (m<8)  at lane m,    acc[m]
    //   lanes 16-31: N=lane-16, vgpr v holds M=v+8    -> diag m (m>=8) at lane m+16, acc[m-8]
    int cidx = lane & 7;
    float v = acc[0];
    v = (cidx == 1) ? acc[1] : v;
    v = (cidx == 2) ? acc[2] : v;
    v = (cidx == 3) ? acc[3] : v;
    v = (cidx == 4) ? acc[4] : v;
    v = (cidx == 5) ? acc[5] : v;
    v = (cidx == 6) ? acc[6] : v;
    v = (cidx == 7) ? acc[7] : v;

    bool diag_owner = (lane < 8) || (lane >= 24);
    int  m  = (lane < 8) ? lane : (lane - 16);
    int  eo = eBase + m;
    if (diag_owner && eo < n_edges) {
        out[eo] = v;
    }
}

extern "C" void kernel_launch(void* const* d_in, const int* in_sizes, int n_in,
                              void* d_out, int out_size, void* d_ws, size_t ws_size,
                              hipStream_t stream) {
    const float* x    = (const float*)d_in[0];   // [N_NODES, 128] f32
    const float* remb = (const float*)d_in[1];   // [N_REL, 128]   f32
    const int*   src  = (const int*)d_in[2];     // [E]
    const int*   dst  = (const int*)d_in[3];     // [E]
    const int*   rel  = (const int*)d_in[4];     // [E]
    float*       out  = (float*)d_out;           // [E] f32

    int n_edges = in_sizes[2];
    int n_rel   = in_sizes[1] / DIM;

    int grid = (n_edges + EDGES_PER_BLOCK - 1) / EDGES_PER_BLOCK;  // 12800
    distmult_wmma_kernel<<<grid, 256, 0, stream>>>(x, remb, src, dst, rel, out,
                                                   n_edges, n_rel);
}